// ScaleULayer_12180527252103
// MI455X (gfx1250) — compile-verified
//
#include <hip/hip_runtime.h>
#include <hip/hip_bf16.h>
#include <math.h>

typedef float v4f __attribute__((ext_vector_type(4)));
typedef int   v4i __attribute__((ext_vector_type(4)));
#define AS1 __attribute__((address_space(1)))
#define AS3 __attribute__((address_space(3)))

#if __has_builtin(__builtin_amdgcn_global_load_async_to_lds_b128)
#define USE_ASYNC_LDS 1
#warning "CDNA5 path: __builtin_amdgcn_global_load_async_to_lds_b128 ACTIVE"
#else
#define USE_ASYNC_LDS 0
#warning "CDNA5 path: async-to-LDS builtin ABSENT, using vmem+ds fallback"
#endif

#if USE_ASYNC_LDS
__device__ __forceinline__ void wait_async0() {
#if __has_builtin(__builtin_amdgcn_s_wait_asynccnt)
  __builtin_amdgcn_s_wait_asynccnt(0);
#else
  asm volatile("s_wait_asynccnt 0" ::: "memory");
#endif
}
#endif

__device__ __forceinline__ float wave_sum32(float v) {
#pragma unroll
  for (int off = 16; off > 0; off >>= 1) v += __shfl_xor(v, off, 32);
  return v;
}

// Grid: (C, 3, B). One 256-thread block per 64x64 output image (HW=4096).
// section 0: out = x * (tanh(b_scaling[c])+1)
// section 1/2: out = fourier_filter(skip, 1, tanh(w)+1)  -- only 4 spectral
//   coeffs are scaled, so filter == x + (t-1)/4096 * 7-term trig expansion.
__global__ __launch_bounds__(256) void scaleu_fourier_kernel(
    const float* __restrict__ x, const float* __restrict__ skip0,
    const float* __restrict__ skip1, const float* __restrict__ b_scaling,
    const float* __restrict__ w0, const float* __restrict__ w1,
    float* __restrict__ out, int C) {
  constexpr int HW = 4096;
  const int c   = blockIdx.x;
  const int sec = blockIdx.y;
  const int b   = blockIdx.z;
  const int tid = threadIdx.x;
  // out image index within (B, 3C, H, W)
  v4f* __restrict__ dst4 =
      (v4f*)(out + ((size_t)(b * 3 + sec) * C + c) * HW);

  __shared__ float tile[4096];
  __shared__ float ctab[64];
  __shared__ float stab[64];
  __shared__ float wred[8 * 8];
  __shared__ float coef[8];

  if (sec == 0) {
    const float bs = tanhf(b_scaling[c]) + 1.0f;
    const v4f* src4 = (const v4f*)(x + (size_t)(b * C + c) * HW);
#pragma unroll
    for (int k = 0; k < 4; ++k) {
      const int i = tid + 256 * k;
      v4f v = __builtin_nontemporal_load(src4 + i);
      __builtin_nontemporal_store(v * bs, dst4 + i);
    }
    return;
  }

  const float* __restrict__ src =
      ((sec == 1) ? skip0 : skip1) + (size_t)(b * C + c) * HW;
  const float t = tanhf(((sec == 1) ? w0 : w1)[0]) + 1.0f;

  // ---- Stage 16KB image into LDS via async DMA; overlap trig-table calc ----
#if USE_ASYNC_LDS
#pragma unroll
  for (int k = 0; k < 4; ++k) {
    const int i = tid + 256 * k;  // float4 index, lane-consecutive 16B
    __builtin_amdgcn_global_load_async_to_lds_b128(
        (AS1 v4i*)(uintptr_t)(src + 4 * i),
        (AS3 v4i*)&tile[4 * i], 0, 0);
  }
#endif
  if (tid < 64) {
    const float ang = (float)tid * 0.0981747704246810387f;  // 2*pi/64
    ctab[tid] = cosf(ang);
    stab[tid] = sinf(ang);
  }
#if USE_ASYNC_LDS
  wait_async0();
#else
  {
    v4f* t4w = (v4f*)tile;
#pragma unroll
    for (int k = 0; k < 4; ++k) {
      const int i = tid + 256 * k;
      t4w[i] = __builtin_nontemporal_load((const v4f*)src + i);
    }
  }
#endif
  __syncthreads();

  // ---- 7 spectral sums: A, B=Σx·e^{iθn}, C=Σx·e^{iθm}, D=Σx·e^{iθ(m+n)} ----
  float A = 0.f, Br = 0.f, Bi = 0.f, Cr = 0.f, Ci = 0.f, Dr = 0.f, Di = 0.f;
  const v4f* t4 = (const v4f*)tile;
#pragma unroll
  for (int k = 0; k < 4; ++k) {
    const int i  = tid + 256 * k;   // element base 4i: h fixed per (thread,k)
    const int h  = i >> 4;
    const int wb = (4 * i) & 63;
    const v4f v  = t4[i];
    const float ch = ctab[h], sh = stab[h];
    float a = 0.f, br = 0.f, bi = 0.f;
#pragma unroll
    for (int j = 0; j < 4; ++j) {
      const float vv = v[j];
      a += vv;
      br += vv * ctab[wb + j];
      bi += vv * stab[wb + j];
    }
    A  += a;  Br += br;  Bi += bi;
    Cr += ch * a;              Ci += sh * a;               // m-weights factor out
    Dr += ch * br - sh * bi;   Di += sh * br + ch * bi;    // angle addition
  }

  A = wave_sum32(A);   Br = wave_sum32(Br); Bi = wave_sum32(Bi);
  Cr = wave_sum32(Cr); Ci = wave_sum32(Ci);
  Dr = wave_sum32(Dr); Di = wave_sum32(Di);

  const int lane = tid & 31, wv = tid >> 5;
  if (lane == 0) {
    float* r = &wred[wv * 8];
    r[0] = A; r[1] = Br; r[2] = Bi; r[3] = Cr; r[4] = Ci; r[5] = Dr; r[6] = Di;
  }
  __syncthreads();
  if (tid < 7) {  // fixed-order cross-wave sum -> bitwise deterministic
    float s = 0.f;
#pragma unroll
    for (int q = 0; q < 8; ++q) s += wred[q * 8 + tid];
    coef[tid] = s * ((t - 1.0f) * (1.0f / 4096.0f));
  }
  __syncthreads();

  const float k0 = coef[0], k1 = coef[1], k2 = coef[2], k3 = coef[3],
              k4 = coef[4], k5 = coef[5], k6 = coef[6];

  // out = x + k0 + k1·cw + k2·sw + k3·ch + k4·sh + k5·cos(h+w) + k6·sin(h+w)
#pragma unroll
  for (int k = 0; k < 4; ++k) {
    const int i  = tid + 256 * k;
    const int h  = i >> 4;
    const int wb = (4 * i) & 63;
    const float ch = ctab[h], sh = stab[h];
    const float K0 = k0 + k3 * ch + k4 * sh;
    const float Kc = k1 + k5 * ch + k6 * sh;
    const float Ks = k2 - k5 * sh + k6 * ch;
    const v4f v = t4[i];
    v4f o;
#pragma unroll
    for (int j = 0; j < 4; ++j)
      o[j] = v[j] + K0 + Kc * ctab[wb + j] + Ks * stab[wb + j];
    __builtin_nontemporal_store(o, dst4 + i);
  }
}

extern "C" void kernel_launch(void* const* d_in, const int* in_sizes, int n_in,
                              void* d_out, int out_size, void* d_ws, size_t ws_size,
                              hipStream_t stream) {
  const float* x  = (const float*)d_in[0];
  const float* s0 = (const float*)d_in[1];
  const float* s1 = (const float*)d_in[2];
  const float* bs = (const float*)d_in[3];
  const float* w0 = (const float*)d_in[4];
  const float* w1 = (const float*)d_in[5];
  float* out = (float*)d_out;

  const int C = in_sizes[3];               // 320
  const int B = in_sizes[0] / (C * 4096);  // 8

  dim3 grid(C, 3, B);                      // x=c, y=section, z=b -> no div/mod
  scaleu_fourier_kernel<<<grid, 256, 0, stream>>>(x, s0, s1, bs, w0, w1, out,
                                                  C);
}